// SimpleAttention_31988916420678
// MI455X (gfx1250) — compile-verified
//
#include <hip/hip_runtime.h>
#include <math.h>

typedef __attribute__((ext_vector_type(2))) float v2f;
typedef __attribute__((ext_vector_type(8))) float v8f;

#define Bz 16
#define Sz 4096
#define Nz 1024
#define Mz 1024
#define EPSF 1e-5f

// ---------------- Pass 1: xsum[b,j] = sum_s x[b,s,j] ----------------
__global__ __launch_bounds__(256) void xsum_kernel(const float* __restrict__ x,
                                                   float* __restrict__ xsum) {
    int b = blockIdx.y;
    int j = blockIdx.x * 256 + threadIdx.x;
    const float* p = x + ((size_t)b * Sz) * Nz + j;
    float acc = 0.f;
    for (int s = 0; s < Sz; ++s) acc += p[(size_t)s * Nz];
    xsum[b * Nz + j] = acc;
}

// ---------------- qraw[b,i] = A[i,:]·xsum[b,:], partial sum of squares ----------------
__global__ __launch_bounds__(128) void query_kernel(const float* __restrict__ A,
                                                    const float* __restrict__ xsum,
                                                    float* __restrict__ qraw,
                                                    float* __restrict__ qsspart) {
    int b = blockIdx.y;
    int i = blockIdx.x * 128 + threadIdx.x;
    const float* ar = A + (size_t)i * Nz;
    const float* xr = xsum + b * Nz;
    float acc = 0.f;
    for (int j = 0; j < Nz; ++j) acc += ar[j] * xr[j];
    qraw[b * Mz + i] = acc;
    __shared__ float red[128];
    red[threadIdx.x] = acc * acc;
    __syncthreads();
    for (int off = 64; off > 0; off >>= 1) {
        if (threadIdx.x < off) red[threadIdx.x] += red[threadIdx.x + off];
        __syncthreads();
    }
    if (threadIdx.x == 0) qsspart[b * 8 + blockIdx.x] = red[0];
}

// ---------------- rnorm[b] = 1/(eps + ||qraw[b]||) ----------------
__global__ void rnorm_kernel(const float* __restrict__ qsspart, float* __restrict__ rnorm) {
    int b = threadIdx.x;  // 16 threads
    float s = 0.f;
    for (int p = 0; p < 8; ++p) s += qsspart[b * 8 + p];
    rnorm[b] = 1.f / (EPSF + sqrtf(s));
}

// ---------------- tvec[b,j] = rnorm[b] * sum_i qraw[b,i]*B[i,j] ----------------
__global__ __launch_bounds__(128) void tvec_kernel(const float* __restrict__ Bm,
                                                   const float* __restrict__ qraw,
                                                   const float* __restrict__ rnorm,
                                                   float* __restrict__ tvec) {
    int b = blockIdx.y;
    int j = blockIdx.x * 128 + threadIdx.x;
    const float* q = qraw + b * Mz;
    float acc = 0.f;
    for (int i = 0; i < Mz; ++i) acc += q[i] * Bm[(size_t)i * Nz + j];
    tvec[b * Nz + j] = acc * rnorm[b];
}

// ---------------- Pass 2 (fused): key-norm GEMM via WMMA + sim + out partial ----------------
// One wave (32 threads) per block; block handles 16 tokens of one batch.
__global__ __launch_bounds__(32) void keysim_kernel(const float* __restrict__ x,
                                                    const float* __restrict__ Bm,
                                                    const float* __restrict__ tvec,
                                                    float* __restrict__ simOut,
                                                    float* __restrict__ outPart) {
    __shared__ float xs[16 * 1024];  // 64KB: 16 tokens x 1024 features

    const int b = blockIdx.y;
    const int chunk = blockIdx.x;      // 256 chunks of 16 tokens
    const int lane = threadIdx.x;      // 0..31 (wave32)

    // Stage x tile into LDS (float4 vectorized)
    {
        const float4* src = (const float4*)(x + (((size_t)b * Sz + (size_t)chunk * 16) * Nz));
        float4* dst = (float4*)xs;
        for (int i = lane; i < (16 * 1024) / 4; i += 32) dst[i] = src[i];
    }
    __syncthreads();

    const int row = lane & 15;          // token (A operand) / B row offset (B operand)
    const int khalf = (lane >> 4) << 1; // 0 or 2: which K-pair this lane holds

    // keynorm2 accumulation: np[r] holds squares for token (r + 8*(lane>=16)), channel (lane&15)
    float np[8] = {0.f, 0.f, 0.f, 0.f, 0.f, 0.f, 0.f, 0.f};
    const float* xr = xs + (row << 10);

    for (int mt = 0; mt < 64; ++mt) {   // 64 tiles of 16 output channels
        const float* Brow = Bm + ((size_t)((mt << 4) + row) << 10);
        v8f acc = {0.f, 0.f, 0.f, 0.f, 0.f, 0.f, 0.f, 0.f};
#pragma unroll 4
        for (int k0 = 0; k0 < 1024; k0 += 4) {
            v2f av = *(const v2f*)(xr + k0 + khalf);     // LDS: A = x tile (16x4)
            v2f bv = *(const v2f*)(Brow + k0 + khalf);   // global/L2: B^T tile (4x16)
            acc = __builtin_amdgcn_wmma_f32_16x16x4_f32(
                false, av, false, bv, (short)0, acc, false, false);
        }
#pragma unroll
        for (int r = 0; r < 8; ++r) np[r] += acc[r] * acc[r];
    }

    // Reduce squares across the 16 channels (lanes within each 16-lane group)
#pragma unroll
    for (int r = 0; r < 8; ++r) {
        float v = np[r];
        v += __shfl_xor(v, 1, 32);
        v += __shfl_xor(v, 2, 32);
        v += __shfl_xor(v, 4, 32);
        v += __shfl_xor(v, 8, 32);
        np[r] = v;  // low half: token r ; high half: token r+8
    }
    float nh[8];
#pragma unroll
    for (int r = 0; r < 8; ++r) nh[r] = __shfl_xor(np[r], 16, 32);  // bring other half over

    // dot[t] = tvec[b,:]·x[t,:] : lane handles token (lane&15), half (lane>>4) of features
    const float* tb = tvec + (size_t)b * Nz;
    float d = 0.f;
    {
        const int t = lane & 15;
        const int h = (lane >> 4) << 9;  // 0 or 512
        const float* xp = xs + (t << 10) + h;
        const float* tp = tb + h;
        for (int j = 0; j < 512; ++j) d += xp[j] * tp[j];
        d += __shfl_xor(d, 16, 32);  // both halves now hold full dot for token (lane&15)
    }

    // sim for token (lane&15), computed in lanes 0..15 (constant-indexed norm select)
    float sim = 0.f;
    if (lane < 16) {
        float n2 = 0.f;
#pragma unroll
        for (int r = 0; r < 8; ++r) {
            n2 = (lane == r) ? np[r] : n2;       // tokens 0..7 live in low-half np
            n2 = (lane == r + 8) ? nh[r] : n2;   // tokens 8..15 came from high half
        }
        sim = d / (EPSF + sqrtf(n2));
        simOut[(size_t)b * Sz + (size_t)chunk * 16 + lane] = sim;
    }

    // out partial: op[j] = sum_t sim[t]*x[t,j]; lane owns j = lane + 32*i
    float oacc[32];
#pragma unroll
    for (int i = 0; i < 32; ++i) oacc[i] = 0.f;
    for (int t = 0; t < 16; ++t) {
        float sv = __shfl(sim, t, 32);  // broadcast sim[t] from lane t
        const float* xp = xs + (t << 10);
#pragma unroll
        for (int i = 0; i < 32; ++i) oacc[i] += sv * xp[lane + (i << 5)];
    }
    float* op = outPart + (((size_t)b * 256 + chunk) * Nz);
#pragma unroll
    for (int i = 0; i < 32; ++i) op[lane + (i << 5)] = oacc[i];
}

// ---------------- Deterministic sequential reduce of out partials ----------------
__global__ __launch_bounds__(256) void outreduce_kernel(const float* __restrict__ outPart,
                                                        float* __restrict__ out) {
    int b = blockIdx.x;
    for (int j = threadIdx.x; j < Nz; j += 256) {
        const float* p = outPart + ((size_t)b * 256) * Nz + j;
        float acc = 0.f;
        for (int c = 0; c < 256; ++c) acc += p[(size_t)c * Nz];
        out[(size_t)b * Nz + j] = acc;
    }
}

extern "C" void kernel_launch(void* const* d_in, const int* in_sizes, int n_in,
                              void* d_out, int out_size, void* d_ws, size_t ws_size,
                              hipStream_t stream) {
    const float* x  = (const float*)d_in[0];   // [16,4096,1024]
    const float* A  = (const float*)d_in[1];   // [1024,1024]
    const float* Bm = (const float*)d_in[2];   // [1024,1024]

    float* out    = (float*)d_out;             // [16,1024]
    float* simOut = out + Bz * Nz;             // [16,4096]

    float* ws      = (float*)d_ws;
    float* xsum    = ws;                        // 16*1024
    float* qraw    = xsum + Bz * Nz;            // 16*1024
    float* qsspart = qraw + Bz * Mz;            // 16*8
    float* rnorm   = qsspart + Bz * 8;          // 16
    float* tvec    = rnorm + Bz;                // 16*1024
    float* outPart = tvec + Bz * Nz;            // 16*256*1024 (16 MB)

    xsum_kernel<<<dim3(4, 16), 256, 0, stream>>>(x, xsum);
    query_kernel<<<dim3(8, 16), 128, 0, stream>>>(A, xsum, qraw, qsspart);
    rnorm_kernel<<<1, 16, 0, stream>>>(qsspart, rnorm);
    tvec_kernel<<<dim3(8, 16), 128, 0, stream>>>(Bm, qraw, rnorm, tvec);
    keysim_kernel<<<dim3(256, 16), 32, 0, stream>>>(x, Bm, tvec, simOut, outPart);
    outreduce_kernel<<<16, 256, 0, stream>>>(outPart, out);
}